// GDSOM_927712936093
// MI455X (gfx1250) — compile-verified
//
#include <hip/hip_runtime.h>
#include <math.h>

#define K_CODES 512
#define D_DIM   256
#define N_VEC   65536              // 64*32*32
#define Q_ELEMS (N_VEC * D_DIM)    // 16777216

typedef __attribute__((ext_vector_type(2))) float v2f;
typedef __attribute__((ext_vector_type(8))) float v8f;
typedef __attribute__((ext_vector_type(4))) unsigned int u32x4;
typedef __attribute__((ext_vector_type(8))) int          i32x8;
typedef __attribute__((ext_vector_type(4))) int          i32x4;

// ---------------------------------------------------------------------------
// Zero the accumulators (counts histogram + mse accumulator).
__global__ void som_init(int* __restrict__ counts, float* __restrict__ mseAcc) {
    int t = threadIdx.x;
    if (t < K_CODES) counts[t] = 0;
    if (t == 0) mseAcc[0] = 0.0f;
}

// ---------------------------------------------------------------------------
// wsq[j] = ||w_j||^2
__global__ void som_wsq(const float* __restrict__ w, float* __restrict__ wsq) {
    int j = blockIdx.x, t = threadIdx.x;
    float v = w[j * D_DIM + t];
    __shared__ float red[256];
    red[t] = v * v;
    __syncthreads();
    for (int o = 128; o > 0; o >>= 1) {
        if (t < o) red[t] += red[t + o];
        __syncthreads();
    }
    if (t == 0) wsq[j] = red[0];
}

// ---------------------------------------------------------------------------
// TDM: DMA one 16-code x 256-dim fp32 w tile (16 KB contiguous) into LDS,
// with hardware padding of 2 DWORDs after every 256 DWORDs -> LDS row stride
// 258 floats (bank-conflict-free B-fragment ds_load_b64 across 16 lanes).
#define WT_STRIDE 258

__device__ __forceinline__ void tdm_load_w_tile(const float* w, int tile,
                                                unsigned int ldsOff) {
    unsigned long long gaddr =
        (unsigned long long)(size_t)w + (unsigned long long)tile * (16u * D_DIM * 4u);
    // D# group0: [1:0]=count=1 | [63:32]=lds_addr | [120:64]=global_addr | [127:126]=type=2
    u32x4 g0 = { 1u,
                 ldsOff,
                 (unsigned int)gaddr,
                 (unsigned int)((gaddr >> 32) & 0x01FFFFFFu) | (2u << 30) };
    // D# group1:
    //  w0: data_size=2(4B)<<16 | pad_enable<<20 | pad_interval=7(256dw)<<22 | pad_amount=1(2dw)<<25
    //  tensor_dim0=131072 (whole w), tensor_dim1=1, tile_dim0=4096 (one linear row),
    //  tile_dim1/2 unused, tensor_dim0_stride=131072
    i32x8 g1 = { (int)0x03D20000,
                 0,                    // barrier addr / tensor_dim0[15:0]=0
                 (int)0x00010002,      // tensor_dim0[31:16]=2 | tensor_dim1 lo16 = 1
                 (int)0x10000000,      // tensor_dim1 hi=0 | tile_dim0=4096 << 16
                 0,                    // tile_dim1=0, tile_dim2=0
                 (int)0x00020000,      // tensor_dim0_stride = 131072
                 0, 0 };
    i32x4 gz4 = { 0, 0, 0, 0 };        // groups 2/3 unused (2D descriptor)
    i32x8 gz8 = { 0, 0, 0, 0, 0, 0, 0, 0 };
    __builtin_amdgcn_tensor_load_to_lds(g0, g1, gz4, gz4, gz8, 0);
}

// ---------------------------------------------------------------------------
// Nearest-code search via fp32 WMMA: per row n, argmin_k (wsq[k] - 2 x_n.w_k).
// 8 waves/block, each wave owns a 16-row slab (128 rows/block).  512 codes in
// 32 tiles of 16, TDM double-buffered through LDS.  K-reduction split over 4
// independent WMMA accumulator chains to run at throughput, not latency.
__global__ __launch_bounds__(256) void som_argmin(
        const float* __restrict__ x, const float* __restrict__ w,
        const float* __restrict__ wsq, int* __restrict__ enc,
        int* __restrict__ counts) {
    __shared__ float wt[2][16 * WT_STRIDE];

    const int t    = threadIdx.x;
    const int wave = t >> 5;
    const int lane = t & 31;
    const int hf   = lane >> 4;     // which K-half of the wave (WMMA layout)
    const int lr   = lane & 15;

    const int rowBase = blockIdx.x * 128 + wave * 16;
    const float* xrow = x + (size_t)(rowBase + lr) * D_DIM;

    // A-matrix (16x4 fp32) fragments for the whole D=256: lane<16 -> K{0,1},
    // lane>=16 -> K{2,3} of each 4-wide K step.  Registers, reused 32x.
    v2f afrag[64];
#pragma unroll
    for (int i = 0; i < 64; ++i)
        afrag[i] = *(const v2f*)(xrow + 4 * i + 2 * hf);

    float runMin[8];
    int   runIdx[8];
#pragma unroll
    for (int r = 0; r < 8; ++r) { runMin[r] = 3.4e38f; runIdx[r] = 0; }

    if (wave == 0)
        tdm_load_w_tile(w, 0, (unsigned int)(size_t)(&wt[0][0]));

    for (int ct = 0; ct < 32; ++ct) {
        if (wave == 0) {
            if (ct + 1 < 32) {
                tdm_load_w_tile(w, ct + 1,
                                (unsigned int)(size_t)(&wt[(ct + 1) & 1][0]));
                __builtin_amdgcn_s_wait_tensorcnt(1);  // oldest (tile ct) landed
            } else {
                __builtin_amdgcn_s_wait_tensorcnt(0);
            }
        }
        __syncthreads();                               // publish tile ct

        const float* wb = &wt[ct & 1][0];
        v8f a0 = {0.f,0.f,0.f,0.f,0.f,0.f,0.f,0.f};
        v8f a1 = {0.f,0.f,0.f,0.f,0.f,0.f,0.f,0.f};
        v8f a2 = {0.f,0.f,0.f,0.f,0.f,0.f,0.f,0.f};
        v8f a3 = {0.f,0.f,0.f,0.f,0.f,0.f,0.f,0.f};
#pragma unroll
        for (int i = 0; i < 64; i += 4) {
            const float* bb = wb + lr * WT_STRIDE + 2 * hf;
            v2f b0 = *(const v2f*)(bb + 4 * (i + 0));
            v2f b1 = *(const v2f*)(bb + 4 * (i + 1));
            v2f b2 = *(const v2f*)(bb + 4 * (i + 2));
            v2f b3 = *(const v2f*)(bb + 4 * (i + 3));
            a0 = __builtin_amdgcn_wmma_f32_16x16x4_f32(false, afrag[i + 0], false, b0,
                                                       (short)0, a0, false, false);
            a1 = __builtin_amdgcn_wmma_f32_16x16x4_f32(false, afrag[i + 1], false, b1,
                                                       (short)0, a1, false, false);
            a2 = __builtin_amdgcn_wmma_f32_16x16x4_f32(false, afrag[i + 2], false, b2,
                                                       (short)0, a2, false, false);
            a3 = __builtin_amdgcn_wmma_f32_16x16x4_f32(false, afrag[i + 3], false, b3,
                                                       (short)0, a3, false, false);
        }

        const int   code = ct * 16 + lr;      // C layout: N = lane%16
        const float wq   = wsq[code];
#pragma unroll
        for (int r = 0; r < 8; ++r) {         // C layout: M = 8*hf + r
            float score = (a0[r] + a1[r]) + (a2[r] + a3[r]);
            float dsc   = wq - 2.0f * score;
            if (dsc < runMin[r]) { runMin[r] = dsc; runIdx[r] = code; }
        }
        __syncthreads();   // all waves done with buf[ct&1] before it is reloaded
    }

    // reduce argmin across the 16 lanes of each half-wave (masks<16 stay in half)
#pragma unroll
    for (int m = 1; m < 16; m <<= 1) {
#pragma unroll
        for (int r = 0; r < 8; ++r) {
            float ov = __shfl_xor(runMin[r], m, 32);
            int   oi = __shfl_xor(runIdx[r], m, 32);
            if (ov < runMin[r] || (ov == runMin[r] && oi < runIdx[r])) {
                runMin[r] = ov; runIdx[r] = oi;
            }
        }
    }

    if (lr == 0) {
        int row0 = rowBase + hf * 8;
#pragma unroll
        for (int r = 0; r < 8; ++r) {
            enc[row0 + r] = runIdx[r];
            atomicAdd(&counts[runIdx[r]], 1);
        }
    }
}

// ---------------------------------------------------------------------------
// quantized = w[enc[n]], accumulate sum((q - x)^2), emit enc as float.
__global__ void som_gather(const float* __restrict__ x, const float* __restrict__ w,
                           const int* __restrict__ enc, float* __restrict__ quant,
                           float* __restrict__ encf, float* __restrict__ mseAcc) {
    int n = blockIdx.x, t = threadIdx.x;
    int e = enc[n];
    float q  = w[(size_t)e * D_DIM + t];
    float xv = x[(size_t)n * D_DIM + t];
    quant[(size_t)n * D_DIM + t] = q;   // scalar store: base is 4B-odd aligned
    float d = q - xv;
    __shared__ float red[256];
    red[t] = d * d;
    __syncthreads();
    for (int o = 128; o > 0; o >>= 1) {
        if (t < o) red[t] += red[t + o];
        __syncthreads();
    }
    if (t == 0) {
        atomicAdd(mseAcc, red[0]);
        encf[n] = (float)e;
    }
}

// ---------------------------------------------------------------------------
// S[j] = sum_k (wsq[k] + wsq[j] - 2 w_j.w_k) * bmat[j,k]   (tiny: 134 MFLOP)
__global__ void som_kohonen_s(const float* __restrict__ w, const float* __restrict__ wsq,
                              const float* __restrict__ bmat, float* __restrict__ S) {
    int j = blockIdx.x, t = threadIdx.x;
    __shared__ float wj[D_DIM];
    wj[t] = w[(size_t)j * D_DIM + t];
    __syncthreads();
    float s = 0.f;
    for (int k = t; k < K_CODES; k += 256) {
        float dot = 0.f;
        for (int d = 0; d < D_DIM; ++d) dot = fmaf(wj[d], w[(size_t)k * D_DIM + d], dot);
        float g = wsq[k] + wsq[j] - 2.0f * dot;
        s += g * bmat[(size_t)j * K_CODES + k];
    }
    __shared__ float red[256];
    red[t] = s;
    __syncthreads();
    for (int o = 128; o > 0; o >>= 1) {
        if (t < o) red[t] += red[t + o];
        __syncthreads();
    }
    if (t == 0) S[j] = red[0];
}

// ---------------------------------------------------------------------------
// loss = 1.25*mse + (1/N) sum_j counts[j]*S[j]; perplexity from counts.
__global__ void som_finalize(const int* __restrict__ counts, const float* __restrict__ S,
                             const float* __restrict__ mseAcc, float* __restrict__ lossOut,
                             float* __restrict__ perpOut) {
    int t = threadIdx.x;   // 512 threads
    float c    = (float)counts[t];
    float invN = 1.0f / (float)N_VEC;
    float koh  = c * S[t] * invN;
    float p    = c * invN;
    float ent  = -p * logf(p + 1e-10f);
    __shared__ float r1[512], r2[512];
    r1[t] = koh; r2[t] = ent;
    __syncthreads();
    for (int o = 256; o > 0; o >>= 1) {
        if (t < o) { r1[t] += r1[t + o]; r2[t] += r2[t + o]; }
        __syncthreads();
    }
    if (t == 0) {
        float mse = mseAcc[0] / (float)Q_ELEMS;
        lossOut[0] = 1.25f * mse + r1[0];   // q_latent + 0.25*e_latent + kohonen
        perpOut[0] = expf(r2[0]);
    }
}

// ---------------------------------------------------------------------------
extern "C" void kernel_launch(void* const* d_in, const int* in_sizes, int n_in,
                              void* d_out, int out_size, void* d_ws, size_t ws_size,
                              hipStream_t stream) {
    const float* x    = (const float*)d_in[0];   // [65536, 256]
    const float* w    = (const float*)d_in[1];   // [512, 256]
    const float* bmat = (const float*)d_in[2];   // [512, 512]

    float* out     = (float*)d_out;
    float* lossOut = out;                        // [0]
    float* quant   = out + 1;                    // [1 .. 1+Q)
    float* perpOut = out + 1 + Q_ELEMS;          // [1+Q]
    float* encf    = out + 2 + Q_ELEMS;          // [2+Q .. 2+Q+N)

    char*  ws      = (char*)d_ws;
    int*   enc     = (int*)ws;                                   // N ints
    int*   counts  = (int*)(ws + (size_t)N_VEC * sizeof(int));   // K ints
    float* wsq     = (float*)(ws + (size_t)N_VEC * sizeof(int) + K_CODES * sizeof(int));
    float* S       = wsq + K_CODES;
    float* mseAcc  = S + K_CODES;

    som_init     <<<1,           512, 0, stream>>>(counts, mseAcc);
    som_wsq      <<<K_CODES,     256, 0, stream>>>(w, wsq);
    som_argmin   <<<N_VEC / 128, 256, 0, stream>>>(x, w, wsq, enc, counts);
    som_gather   <<<N_VEC,       256, 0, stream>>>(x, w, enc, quant, encf, mseAcc);
    som_kohonen_s<<<K_CODES,     256, 0, stream>>>(w, wsq, bmat, S);
    som_finalize <<<1,           512, 0, stream>>>(counts, S, mseAcc, lossOut, perpOut);
}